// LIFNR_49134425866321
// MI455X (gfx1250) — compile-verified
//
#include <hip/hip_runtime.h>

// LIF (no-reset) temporal scan recast as H = L * X with L a constant 32x32
// lower-triangular decay matrix (L[t][s] = 0.5^(t-s+1) for s<=t), executed
// with V_WMMA_F32_16X16X4_F32 on gfx1250. Memory-bound: each input element
// is loaded exactly once, each output stored once (non-temporal hints).
// Each wave sweeps BPW column blocks so the lane-dependent A-fragment setup
// (decay coefficients) is amortized across 32 KB of traffic.

typedef float v2f __attribute__((ext_vector_type(2)));
typedef float v8f __attribute__((ext_vector_type(8)));

#define LIF_T     32
#define LIF_COLS  (64 * 32768)   // B*N = 2,097,152 columns
#define BPW       8              // 16-column blocks per wave

// decay coefficient 0.5^(t-s+1) for s<=t, else 0  (exact powers of two)
__device__ __forceinline__ float lif_coef(int t, int s) {
    if (s > t) return 0.0f;
    unsigned e = (unsigned)(127 - (t - s + 1));   // t-s+1 in [1,32] -> normal
    return __uint_as_float(e << 23);
}

__global__ void __launch_bounds__(256)
lif_wmma_kernel(const float* __restrict__ x, float* __restrict__ out) {
    const int lane  = threadIdx.x & 31;
    const int wave  = threadIdx.x >> 5;
    const int wid   = blockIdx.x * (blockDim.x >> 5) + wave;  // global wave id
    const int n     = lane & 15;        // column within tile / M row for A
    const int khalf = lane >> 4;        // 0: K base 0, 1: K base 2

    // ---- Precompute A fragments once per wave (loop-invariant) ----
    // A 16x4 f32 layout: lane (M + 16*(K>=2)), vgpr j holds K = 2*khalf + j.
    float a0[4][2];                     // tile0: rows t = 0..15, chunks k=0..3
    float a1[8][2];                     // tile1: rows t = 16..31, chunks k=0..7
#pragma unroll
    for (int kc = 0; kc < 4; ++kc)
#pragma unroll
        for (int j = 0; j < 2; ++j)
            a0[kc][j] = lif_coef(n, 4 * kc + 2 * khalf + j);
#pragma unroll
    for (int kc = 0; kc < 8; ++kc)
#pragma unroll
        for (int j = 0; j < 2; ++j)
            a1[kc][j] = lif_coef(16 + n, 4 * kc + 2 * khalf + j);

    // ---- Sweep BPW consecutive 16-column blocks ----
    const int c_base = wid * (16 * BPW);
#pragma unroll 1
    for (int blk = 0; blk < BPW; ++blk) {
        const int c0 = c_base + blk * 16;

        v8f acc0 = {};
        v8f acc1 = {};

        // K loop: stream X rows once, feed both output tiles
        const float* xcol = x + c0 + n;
#pragma unroll
        for (int kc = 0; kc < 8; ++kc) {
            const int s0 = 4 * kc + 2 * khalf;
            // B 4x16 f32 layout (mirror of A): lanes 0-15 K base 0, 16-31 K base 2
            v2f b;
            b.x = __builtin_nontemporal_load(xcol + (size_t)s0 * LIF_COLS);
            b.y = __builtin_nontemporal_load(xcol + (size_t)(s0 + 1) * LIF_COLS);

            if (kc < 4) {   // L is lower-triangular: tile0 only needs s < 16
                v2f a = { a0[kc][0], a0[kc][1] };
                acc0 = __builtin_amdgcn_wmma_f32_16x16x4_f32(
                    false, a, false, b, (short)0, acc0, false, false);
            }
            v2f a2 = { a1[kc][0], a1[kc][1] };
            acc1 = __builtin_amdgcn_wmma_f32_16x16x4_f32(
                false, a2, false, b, (short)0, acc1, false, false);
        }

        // Threshold + store.
        // C/D f32 16x16 layout: VGPR r -> row (r + 8*khalf), col n.
        float* ocol = out + c0 + n;
#pragma unroll
        for (int r = 0; r < 8; ++r) {
            const int t0 = r + 8 * khalf;
            float s0v = (acc0[r] >= 1.0f) ? 1.0f : 0.0f;
            float s1v = (acc1[r] >= 1.0f) ? 1.0f : 0.0f;
            __builtin_nontemporal_store(s0v, ocol + (size_t)t0 * LIF_COLS);
            __builtin_nontemporal_store(s1v, ocol + (size_t)(16 + t0) * LIF_COLS);
        }
    }
}

extern "C" void kernel_launch(void* const* d_in, const int* in_sizes, int n_in,
                              void* d_out, int out_size, void* d_ws, size_t ws_size,
                              hipStream_t stream) {
    const float* x = (const float*)d_in[0];
    float* out = (float*)d_out;
    (void)in_sizes; (void)n_in; (void)out_size; (void)d_ws; (void)ws_size;

    const int col_blocks      = LIF_COLS / 16;                    // 131,072
    const int waves           = col_blocks / BPW;                 // 16,384
    const int waves_per_block = 8;                                // 256 thr
    const int grid            = waves / waves_per_block;          // 2,048
    hipLaunchKernelGGL(lif_wmma_kernel, dim3(grid), dim3(256), 0, stream, x, out);
}